// DGC_50216757625453
// MI455X (gfx1250) — compile-verified
//
#include <hip/hip_runtime.h>
#include <hip/hip_bf16.h>

// Problem constants (match reference)
#define NN      100000
#define EE      1600000
#define DD      128
#define CC      40
#define KSTEPS  4
#define DELTA   0.5f
#define LN_EPS  1e-5f

typedef __attribute__((ext_vector_type(2))) float v2f;
typedef __attribute__((ext_vector_type(8))) float v8f;

// ---------------------------------------------------------------------------
// Degree / normalization setup
// ---------------------------------------------------------------------------
__global__ void dgc_deg_init(float* __restrict__ deg) {
    int i = blockIdx.x * blockDim.x + threadIdx.x;
    if (i < NN) deg[i] = 1.0f;                 // self loop
}

__global__ void dgc_deg_count(const int* __restrict__ row, float* __restrict__ deg) {
    int e = blockIdx.x * blockDim.x + threadIdx.x;
    if (e < EE) atomicAdd(&deg[row[e]], 1.0f);
}

__global__ void dgc_deg_rsqrt(float* __restrict__ deg) {
    int i = blockIdx.x * blockDim.x + threadIdx.x;
    if (i < NN) deg[i] = rsqrtf(deg[i]);       // deg buffer becomes dinv
}

// ---------------------------------------------------------------------------
// Propagation step: agg = src * w_self  (w_self = dinv^2)
// ---------------------------------------------------------------------------
__global__ void dgc_init_agg(const float* __restrict__ src,
                             const float* __restrict__ dinv,
                             float* __restrict__ agg) {
    int idx = blockIdx.x * blockDim.x + threadIdx.x;   // one float4 per thread
    if (idx >= NN * (DD / 4)) return;
    int node = idx / (DD / 4);
    float ws = dinv[node] * dinv[node];
    const float4* s = (const float4*)src;
    float4 v = s[idx];
    v.x *= ws; v.y *= ws; v.z *= ws; v.w *= ws;
    ((float4*)agg)[idx] = v;
}

// agg[row] += src[col] * dinv[row]*dinv[col]   (one wave per edge; 4 f32
// atomics per lane; working set (~103MB) lives in the 192MB L2)
__global__ void dgc_edge_scatter(const float* __restrict__ src,
                                 const int* __restrict__ row,
                                 const int* __restrict__ col,
                                 const float* __restrict__ dinv,
                                 float* __restrict__ agg) {
    int e    = (blockIdx.x * blockDim.x + threadIdx.x) >> 5;
    int lane = threadIdx.x & 31;
    if (e >= EE) return;
    int r = row[e];
    int c = col[e];
    float w = dinv[r] * dinv[c];
    float4 v = ((const float4*)(src + (size_t)c * DD))[lane];
    float* dst = agg + (size_t)r * DD + lane * 4;
    atomicAdd(dst + 0, v.x * w);
    atomicAdd(dst + 1, v.y * w);
    atomicAdd(dst + 2, v.z * w);
    atomicAdd(dst + 3, v.w * w);
}

// h = (1-delta)*src + delta*agg
__global__ void dgc_combine(const float* __restrict__ src,
                            const float* __restrict__ agg,
                            float* __restrict__ h) {
    int idx = blockIdx.x * blockDim.x + threadIdx.x;
    if (idx >= NN * (DD / 4)) return;
    float4 a = ((const float4*)src)[idx];
    float4 g = ((const float4*)agg)[idx];
    float4 o;
    o.x = (1.0f - DELTA) * a.x + DELTA * g.x;
    o.y = (1.0f - DELTA) * a.y + DELTA * g.y;
    o.z = (1.0f - DELTA) * a.z + DELTA * g.z;
    o.w = (1.0f - DELTA) * a.w + DELTA * g.w;
    ((float4*)h)[idx] = o;
}

// ---------------------------------------------------------------------------
// FC: out[N,40] = h[N,128] @ W^T + b   via V_WMMA_F32_16X16X4_F32
// Block stages a zero-padded 48x128 W tile into LDS (24KB) once; inner K loop
// is branch-free: global_load_b64 (A) + ds_load_b64 (B) + chained WMMA.
// One wave per 16(M)x16(N) output tile; K=128 -> 32 WMMAs.
// ---------------------------------------------------------------------------
#define M_TILES   (NN / 16)     // 6250, exact
#define N_TILES   3             // cols 0..47, zero-padded past 40
#define NPAD      (N_TILES * 16)
#define FC_BLOCK  256           // 8 waves per block

__global__ void dgc_fc_wmma(const float* __restrict__ h,
                            const float* __restrict__ W,
                            const float* __restrict__ bias,
                            float* __restrict__ out) {
    __shared__ float wlds[NPAD * DD];          // 48*128*4B = 24KB

    // --- cooperative zero-padded W stage (uniform across the block) ---
    for (int idx = threadIdx.x; idx < NPAD * DD; idx += FC_BLOCK) {
        int colc = idx >> 7;                   // idx / 128
        wlds[idx] = (colc < CC) ? W[idx] : 0.0f;   // W is row-major [40][128]
    }
    __syncthreads();

    int wave = (blockIdx.x * FC_BLOCK + threadIdx.x) >> 5;   // global wave id
    int lane = threadIdx.x & 31;
    int m_tile = wave / N_TILES;
    int n_tile = wave % N_TILES;
    if (m_tile >= M_TILES) return;             // wave-uniform exit

    const int m_base = m_tile * 16;
    const int l16    = lane & 15;
    const int kb     = (lane >> 4) * 2;        // k-pair owned by this lane

    const int mrow   = m_base + l16;           // A row for this lane
    const int ncol_l = n_tile * 16 + l16;      // padded B column (0..47)
    const float* arow = h + (size_t)mrow * DD;
    const float* brow = wlds + ncol_l * DD;

    v8f acc = {};
    #pragma unroll 8
    for (int k = 0; k < DD; k += 4) {
        v2f a, b;
        a.x = arow[k + kb];
        a.y = arow[k + kb + 1];
        b.x = brow[k + kb];
        b.y = brow[k + kb + 1];
        acc = __builtin_amdgcn_wmma_f32_16x16x4_f32(
            /*neg_a=*/false, a, /*neg_b=*/false, b,
            /*c_mod=*/(short)0, acc, /*reuse_a=*/false, /*reuse_b=*/false);
    }

    // D layout: VGPR v -> M = v + 8*(lane/16), N = lane%16
    if (ncol_l < CC) {
        float bv = bias[ncol_l];
        int m_off = (lane >> 4) * 8;
        #pragma unroll
        for (int v = 0; v < 8; ++v) {
            out[(size_t)(m_base + m_off + v) * CC + ncol_l] = acc[v] + bv;
        }
    }
}

// ---------------------------------------------------------------------------
// LayerNorm over C=40, in place on out. One wave per node.
// ---------------------------------------------------------------------------
__global__ void dgc_layernorm(float* __restrict__ out,
                              const float* __restrict__ gamma,
                              const float* __restrict__ beta) {
    int node = (blockIdx.x * blockDim.x + threadIdx.x) >> 5;
    int lane = threadIdx.x & 31;
    if (node >= NN) return;
    float* rowp = out + (size_t)node * CC;

    float x0 = rowp[lane];                               // lanes 0..31
    float x1 = (lane < CC - 32) ? rowp[32 + lane] : 0.f; // lanes 0..7
    float s  = x0 + x1;
    float sq = x0 * x0 + x1 * x1;
    #pragma unroll
    for (int off = 16; off > 0; off >>= 1) {
        s  += __shfl_xor(s,  off, 32);
        sq += __shfl_xor(sq, off, 32);
    }
    float mean = s * (1.0f / CC);
    float var  = sq * (1.0f / CC) - mean * mean;
    float inv  = rsqrtf(var + LN_EPS);

    rowp[lane] = (x0 - mean) * inv * gamma[lane] + beta[lane];
    if (lane < CC - 32)
        rowp[32 + lane] = (x1 - mean) * inv * gamma[32 + lane] + beta[32 + lane];
}

// ---------------------------------------------------------------------------
// Host launcher
// ---------------------------------------------------------------------------
extern "C" void kernel_launch(void* const* d_in, const int* in_sizes, int n_in,
                              void* d_out, int out_size, void* d_ws, size_t ws_size,
                              hipStream_t stream) {
    const float* feat  = (const float*)d_in[0];
    const int*   row   = (const int*)  d_in[1];
    const int*   col   = (const int*)  d_in[2];
    const float* W     = (const float*)d_in[3];
    const float* bias  = (const float*)d_in[4];
    const float* gamma = (const float*)d_in[5];
    const float* beta  = (const float*)d_in[6];
    float* out = (float*)d_out;

    // Workspace carve-up (~103 MB): dinv | h | agg
    const size_t NP = 100352;                    // N padded for alignment
    float* dinv = (float*)d_ws;
    float* h    = dinv + NP;
    float* agg  = h + (size_t)NN * DD;

    const int B = 256;

    // --- degrees -> dinv ---
    dgc_deg_init <<<(NN + B - 1) / B, B, 0, stream>>>(dinv);
    dgc_deg_count<<<(EE + B - 1) / B, B, 0, stream>>>(row, dinv);
    dgc_deg_rsqrt<<<(NN + B - 1) / B, B, 0, stream>>>(dinv);

    // --- K propagation steps (src = feat on step 0, then h in place) ---
    const int elem_blocks = (NN * (DD / 4) + B - 1) / B;   // 12500
    const int edge_blocks = (EE * 32 + B - 1) / B;         // 200000
    for (int step = 0; step < KSTEPS; ++step) {
        const float* src = (step == 0) ? feat : h;
        dgc_init_agg    <<<elem_blocks, B, 0, stream>>>(src, dinv, agg);
        dgc_edge_scatter<<<edge_blocks, B, 0, stream>>>(src, row, col, dinv, agg);
        dgc_combine     <<<elem_blocks, B, 0, stream>>>(src, agg, h);
    }

    // --- FC via f32 WMMA (LDS-staged, zero-padded B) ---
    const int fc_waves  = M_TILES * N_TILES;               // 18750
    const int fc_blocks = (fc_waves * 32 + FC_BLOCK - 1) / FC_BLOCK;
    dgc_fc_wmma<<<fc_blocks, FC_BLOCK, 0, stream>>>(h, W, bias, out);

    // --- LayerNorm in place on out ---
    dgc_layernorm<<<(NN * 32 + B - 1) / B, B, 0, stream>>>(out, gamma, beta);
}